// MRDualHFDNetLayer_78194174591128
// MI455X (gfx1250) — compile-verified
//
#include <hip/hip_runtime.h>
#include <hip/hip_bf16.h>
#include <math.h>

// ---------------------------------------------------------------------------
// MRDualHFDNet layer for MI455X (gfx1250, wave32).
//
// Roofline: bi_adj+adj_mask stream = 512MB @ 23.3TB/s ~= 22us -> HBM floor.
// GEMMs (~2.3 GFLOP fp32) use V_WMMA_F32_16X16X4_F32 (exact fp32 accum).
// Edge phase gathers/atomics hit L2 (h buffers ~8MB/rel, hcat 6MB << 192MB L2).
// Labels staged to LDS via gfx1250 async-to-LDS (ASYNCcnt) so the streaming
// loop issues only 2 VMEM loads + 2 ds_load_b128 per 32 streamed bytes.
// ---------------------------------------------------------------------------

#define N_NODES 8192
#define IN_F    128
#define D_F     256          // OUT*HEAD
#define R_REL   3
#define HEADS   4
#define HD      64
#define E_EDGES 262144
#define DTOT    1024         // [ sproj | h0 | h1 | h2 ] columns
#define OUT_F   64
#define HCOLS   (R_REL * HD) // 192

typedef __attribute__((ext_vector_type(2))) float v2f;
typedef __attribute__((ext_vector_type(4))) float f4;
typedef __attribute__((ext_vector_type(8))) float v8f;
typedef __attribute__((ext_vector_type(4))) int   v4i;

typedef __attribute__((address_space(1))) v4i* gv4i_p;   // global v4i*
typedef __attribute__((address_space(3))) v4i* lv4i_p;   // LDS v4i*

#if defined(__has_builtin)
#if __has_builtin(__builtin_amdgcn_global_load_async_to_lds_b128) && \
    __has_builtin(__builtin_amdgcn_s_wait_asynccnt)
#define USE_ASYNC_LDS 1
#endif
#endif

// ---------------------------------------------------------------------------
// fp32 WMMA GEMM: P[:,0:256)=X@Wd+bd, P[:,256+256r)=X@Ww[r]+bw[r]
// One wave per 16x16 output tile, K stepped by 4 (V_WMMA_F32_16X16X4_F32).
// A layout (16x4 f32): lane m=lane&15, group g=lane>>4 -> K = {2g, 2g+1}.
// B layout (4x16 f32): n=lane&15, VGPR0 holds K=2g row, VGPR1 K=2g+1.
// C/D (16x16 f32): VGPR v -> row v + 8g, col n.
// ---------------------------------------------------------------------------
__global__ void proj_gemm_kernel(const float* __restrict__ X,
                                 const float* __restrict__ Wd,
                                 const float* __restrict__ bd,
                                 const float* __restrict__ Ww,
                                 const float* __restrict__ bw,
                                 float* __restrict__ P) {
  const int lane = threadIdx.x & 31;
  const int wave = threadIdx.x >> 5;
  const int row0 = blockIdx.y << 4;
  const int col0 = (blockIdx.x << 6) + (wave << 4);   // 4 waves x 16 cols
  const int seg  = col0 >> 8;                          // 0=Wd, 1..3=Ww[seg-1]
  const float* W    = (seg == 0) ? Wd : (Ww + (size_t)(seg - 1) * IN_F * D_F);
  const float* bias = (seg == 0) ? bd : (bw + (seg - 1) * D_F);
  const int wcol = col0 & (D_F - 1);

  const int m = lane & 15;
  const int g = lane >> 4;
  const float* aRow = X + (size_t)(row0 + m) * IN_F;

  v8f acc = {};
#pragma unroll 4
  for (int k = 0; k < IN_F; k += 4) {
    v2f a, b;
    a.x = aRow[k + 2 * g + 0];
    a.y = aRow[k + 2 * g + 1];
    const float* bp = W + (size_t)(k + 2 * g) * D_F + wcol + m;
    b.x = bp[0];
    b.y = bp[D_F];
    acc = __builtin_amdgcn_wmma_f32_16x16x4_f32(false, a, false, b,
                                                (short)0, acc, false, false);
  }
  const float bv = bias[wcol + m];
#pragma unroll
  for (int v = 0; v < 8; ++v) {
    const int row = row0 + v + g * 8;
    P[(size_t)row * DTOT + col0 + m] = acc[v] + bv;
  }
}

// ---------------------------------------------------------------------------
// Final GEMM: Hout = hcat(8192x192) @ Wl(192x64) + bl. Same WMMA scheme.
// ---------------------------------------------------------------------------
__global__ void out_gemm_kernel(const float* __restrict__ Hcat,
                                const float* __restrict__ Wl,
                                const float* __restrict__ bl,
                                float* __restrict__ Hout) {
  const int lane = threadIdx.x & 31;
  const int wave = threadIdx.x >> 5;
  const int row0 = blockIdx.x << 4;
  const int col0 = wave << 4;                // 4 waves cover all 64 cols
  const int m = lane & 15;
  const int g = lane >> 4;
  const float* aRow = Hcat + (size_t)(row0 + m) * HCOLS;

  v8f acc = {};
#pragma unroll 4
  for (int k = 0; k < HCOLS; k += 4) {
    v2f a, b;
    a.x = aRow[k + 2 * g + 0];
    a.y = aRow[k + 2 * g + 1];
    const float* bp = Wl + (size_t)(k + 2 * g) * OUT_F + col0 + m;
    b.x = bp[0];
    b.y = bp[OUT_F];
    acc = __builtin_amdgcn_wmma_f32_16x16x4_f32(false, a, false, b,
                                                (short)0, acc, false, false);
  }
  const float bv = bl[col0 + m];
#pragma unroll
  for (int v = 0; v < 8; ++v) {
    const int row = row0 + v + g * 8;
    Hout[(size_t)row * OUT_F + col0 + m] = acc[v] + bv;
  }
}

// ---------------------------------------------------------------------------
// Per-node scalars: p_src/p_dst (256-dot with Wf combos), q_src/q_dst
// (per-relation, per-head 64-dots with Wa). One wave per node.
// ---------------------------------------------------------------------------
__global__ void pq_kernel(const float* __restrict__ P,
                          const float* __restrict__ Wf,
                          const float* __restrict__ Wa,
                          float* __restrict__ p_src, float* __restrict__ p_dst,
                          float* __restrict__ q_src, float* __restrict__ q_dst) {
  const int lane = threadIdx.x & 31;
  const int node = blockIdx.x * 8 + (threadIdx.x >> 5);
  const float* sp = P + (size_t)node * DTOT;

  float ps = 0.f, pd = 0.f;
#pragma unroll
  for (int t = 0; t < 8; ++t) {
    const int j = lane + 32 * t;
    const float w1 = Wf[j], w2 = Wf[D_F + j], w3 = Wf[2 * D_F + j];
    const float v = sp[j];
    ps += v * (w1 + w3);
    pd += v * (w2 - w3);
  }
#pragma unroll
  for (int off = 16; off; off >>= 1) {
    ps += __shfl_xor(ps, off, 32);
    pd += __shfl_xor(pd, off, 32);
  }
  if (lane == 0) { p_src[node] = ps; p_dst[node] = pd; }

  for (int r = 0; r < R_REL; ++r) {
    const float* hrow = P + (size_t)node * DTOT + D_F * (r + 1);
    const float* wa = Wa + r * (2 * HD);
    float qs[HEADS] = {0.f, 0.f, 0.f, 0.f};
    float qd[HEADS] = {0.f, 0.f, 0.f, 0.f};
#pragma unroll
    for (int t = 0; t < 8; ++t) {
      const int j = lane + 32 * t;
      const int hh = t >> 1;          // head = j>>6 for lane<32
      const int jj = j & (HD - 1);
      const float v = hrow[j];
      qs[hh] += v * wa[jj];
      qd[hh] += v * wa[HD + jj];
    }
#pragma unroll
    for (int hh = 0; hh < HEADS; ++hh) {
      float a = qs[hh], b = qd[hh];
#pragma unroll
      for (int off = 16; off; off >>= 1) {
        a += __shfl_xor(a, off, 32);
        b += __shfl_xor(b, off, 32);
      }
      if (lane == 0) {
        q_src[((size_t)r * N_NODES + node) * HEADS + hh] = a;
        q_dst[((size_t)r * N_NODES + node) * HEADS + hh] = b;
      }
    }
  }
}

// ---------------------------------------------------------------------------
// Helpers / small kernels
// ---------------------------------------------------------------------------
__device__ __forceinline__ void atomicMaxFloat(float* addr, float value) {
  // Monotone float-as-int trick -> global_atomic_max_i32 / _min_u32.
  if (value >= 0.f)
    atomicMax((int*)addr, __float_as_int(value));
  else
    atomicMin((unsigned int*)addr, __float_as_uint(value));
}

__global__ void zero_kernel(float* __restrict__ p, int n) {
  const int i = blockIdx.x * blockDim.x + threadIdx.x;
  if (i < n) p[i] = 0.f;
}

__global__ void init_md_kernel(float* __restrict__ m, float* __restrict__ denom) {
  const int i = blockIdx.x * blockDim.x + threadIdx.x;
  m[i] = -INFINITY;
  denom[i] = 0.f;
}

__global__ void m_fix_kernel(float* __restrict__ m) {
  const int i = blockIdx.x * blockDim.x + threadIdx.x;
  const float v = m[i];
  m[i] = __builtin_isfinite(v) ? v : 0.f;
}

// ---------------------------------------------------------------------------
// Edge pass 1: sign + alpha (leaky_relu) + segment-max via atomics.
// q gathers and alpha stores are 128-bit (HEADS=4 contiguous).
// ---------------------------------------------------------------------------
__global__ void edge_alpha_kernel(const int* __restrict__ src,
                                  const int* __restrict__ dst,
                                  const float* __restrict__ p_src,
                                  const float* __restrict__ p_dst,
                                  const float* __restrict__ q_src,
                                  const float* __restrict__ q_dst,
                                  const float* __restrict__ bf,
                                  const float* __restrict__ ba_r,
                                  float* __restrict__ signb,
                                  float* __restrict__ alphab,
                                  float* __restrict__ mbuf) {
  const int e = blockIdx.x * blockDim.x + threadIdx.x;
  const int s = src[e], d = dst[e];
  const float sign = tanhf(p_src[s] + p_dst[d] + bf[0]);
  signb[e] = sign;
  const float bar = ba_r[0];
  const f4 qs = *(const f4*)&q_src[(size_t)s * HEADS];
  const f4 qd = *(const f4*)&q_dst[(size_t)d * HEADS];
  f4 a = sign * qs + qd + bar;
  f4 al;
  al.x = (a.x >= 0.f) ? a.x : 0.01f * a.x;     // leaky_relu, slope 0.01
  al.y = (a.y >= 0.f) ? a.y : 0.01f * a.y;
  al.z = (a.z >= 0.f) ? a.z : 0.01f * a.z;
  al.w = (a.w >= 0.f) ? a.w : 0.01f * a.w;
  *(f4*)&alphab[(size_t)e * HEADS] = al;
  atomicMaxFloat(&mbuf[(size_t)d * HEADS + 0], al.x);
  atomicMaxFloat(&mbuf[(size_t)d * HEADS + 1], al.y);
  atomicMaxFloat(&mbuf[(size_t)d * HEADS + 2], al.z);
  atomicMaxFloat(&mbuf[(size_t)d * HEADS + 3], al.w);
}

// ---------------------------------------------------------------------------
// Edge pass 2: ew = exp(alpha - m[dst]); segment-sum denom. 128-bit accesses.
// ---------------------------------------------------------------------------
__global__ void edge_exp_kernel(const int* __restrict__ dst,
                                float* __restrict__ ewb,     // in: alpha, out: ew
                                const float* __restrict__ mbuf,
                                float* __restrict__ denom) {
  const int e = blockIdx.x * blockDim.x + threadIdx.x;
  const int d = dst[e];
  const f4 a = *(const f4*)&ewb[(size_t)e * HEADS];
  const f4 m = *(const f4*)&mbuf[(size_t)d * HEADS];
  f4 ew;
  ew.x = expf(a.x - m.x);
  ew.y = expf(a.y - m.y);
  ew.z = expf(a.z - m.z);
  ew.w = expf(a.w - m.w);
  *(f4*)&ewb[(size_t)e * HEADS] = ew;
  atomicAdd(&denom[(size_t)d * HEADS + 0], ew.x);
  atomicAdd(&denom[(size_t)d * HEADS + 1], ew.y);
  atomicAdd(&denom[(size_t)d * HEADS + 2], ew.z);
  atomicAdd(&denom[(size_t)d * HEADS + 3], ew.w);
}

// ---------------------------------------------------------------------------
// Edge pass 3: one wave32 per edge. out[dst,k] += sign * sum_h attn_h*h[src,h,k]
// (heads pre-summed in registers -> 64 atomics/edge, all L2-resident).
// ---------------------------------------------------------------------------
__global__ void edge_scatter_kernel(const int* __restrict__ src,
                                    const int* __restrict__ dst,
                                    const float* __restrict__ signb,
                                    const float* __restrict__ ewb,
                                    const float* __restrict__ denom,
                                    const float* __restrict__ P,
                                    int hoff,
                                    float* __restrict__ hcat,
                                    int rcol) {
  const int e = (blockIdx.x * blockDim.x + threadIdx.x) >> 5;
  const int lane = threadIdx.x & 31;
  const int s = src[e], d = dst[e];
  const float sign = signb[e];
  const f4 ew  = *(const f4*)&ewb[(size_t)e * HEADS];
  const f4 den = *(const f4*)&denom[(size_t)d * HEADS];
  const f4 w = ew / den;
  const float* hrow = P + (size_t)s * DTOT + hoff;
  float* orow = hcat + (size_t)d * HCOLS + rcol;
#pragma unroll
  for (int kk = 0; kk < 2; ++kk) {
    const int k = lane + 32 * kk;
    const float v = w.x * hrow[k] + w.y * hrow[HD + k] +
                    w.z * hrow[2 * HD + k] + w.w * hrow[3 * HD + k];
    atomicAdd(&orow[k], sign * v);
  }
}

// ---------------------------------------------------------------------------
// Fused T row-normalize + T@labels. HBM-dominant: 512MB streamed once ->
// non-temporal 128b loads, labels staged to LDS (async-to-LDS when available)
// so the hot loop is 2 VMEM + 2 DS loads per 32 streamed bytes.
// One 8-wave block per row; 64KB dynamic LDS (WGP has 320KB).
// ---------------------------------------------------------------------------
__global__ void y_hat_kernel(const float* __restrict__ bi_adj,
                             const float* __restrict__ adj_mask,
                             const float* __restrict__ labels,
                             float* __restrict__ y_hat) {
  extern __shared__ float lab[];                 // 2*N_NODES floats = 64KB
  const int row = blockIdx.x;
  const float* arow = bi_adj + (size_t)row * N_NODES;
  const float* mrow = adj_mask + (size_t)row * N_NODES;

  // Stage labels (N,2) interleaved into LDS once per block.
#if defined(USE_ASYNC_LDS)
  for (int i4 = threadIdx.x; i4 < 2 * N_NODES / 4; i4 += blockDim.x) {
    __builtin_amdgcn_global_load_async_to_lds_b128(
        (gv4i_p)(labels + 4 * i4),
        (lv4i_p)(lab + 4 * i4), 0, 0);
  }
  __builtin_amdgcn_s_wait_asynccnt(0);
  __syncthreads();
#else
  for (int i4 = threadIdx.x; i4 < 2 * N_NODES / 4; i4 += blockDim.x) {
    *(f4*)&lab[4 * i4] = *(const f4*)&labels[4 * i4];
  }
  __syncthreads();
#endif

  float sabs = 0.f, y0 = 0.f, y1 = 0.f;
  for (int j4 = threadIdx.x; j4 < N_NODES / 4; j4 += blockDim.x) {
    const int j = 4 * j4;
    __builtin_prefetch(arow + j + 4 * blockDim.x, 0, 0);
    __builtin_prefetch(mrow + j + 4 * blockDim.x, 0, 0);
    const f4 a = __builtin_nontemporal_load((const f4*)(arow + j));
    const f4 b = __builtin_nontemporal_load((const f4*)(mrow + j));
    const f4 t = a * b;
    const f4 la = *(const f4*)&lab[2 * j];       // labels j, j+1
    const f4 lb = *(const f4*)&lab[2 * j + 4];   // labels j+2, j+3
    sabs += fabsf(t.x) + fabsf(t.y) + fabsf(t.z) + fabsf(t.w);
    y0 += t.x * la.x + t.y * la.z + t.z * lb.x + t.w * lb.z;
    y1 += t.x * la.y + t.y * la.w + t.z * lb.y + t.w * lb.w;
  }
#pragma unroll
  for (int off = 16; off; off >>= 1) {
    sabs += __shfl_xor(sabs, off, 32);
    y0   += __shfl_xor(y0, off, 32);
    y1   += __shfl_xor(y1, off, 32);
  }
  __shared__ float sred[8], y0red[8], y1red[8];
  const int w = threadIdx.x >> 5;
  if ((threadIdx.x & 31) == 0) { sred[w] = sabs; y0red[w] = y0; y1red[w] = y1; }
  __syncthreads();
  if (threadIdx.x == 0) {
    float s = 0.f, a = 0.f, b = 0.f;
#pragma unroll
    for (int i = 0; i < 8; ++i) { s += sred[i]; a += y0red[i]; b += y1red[i]; }
    s = fmaxf(s, 1e-12f);
    y_hat[(size_t)row * 2 + 0] = a / s;
    y_hat[(size_t)row * 2 + 1] = b / s;
  }
}

// ---------------------------------------------------------------------------
// Host-side orchestration.
// ---------------------------------------------------------------------------
extern "C" void kernel_launch(void* const* d_in, const int* in_sizes, int n_in,
                              void* d_out, int out_size, void* d_ws, size_t ws_size,
                              hipStream_t stream) {
  const float* features = (const float*)d_in[0];
  const float* labels   = (const float*)d_in[1];
  const float* bi_adj   = (const float*)d_in[2];
  const float* adj_mask = (const float*)d_in[3];
  const float* Wd       = (const float*)d_in[4];
  const float* bd       = (const float*)d_in[5];
  const float* Wf       = (const float*)d_in[6];
  const float* bf       = (const float*)d_in[7];
  const float* Ww       = (const float*)d_in[8];
  const float* bw       = (const float*)d_in[9];
  const float* Wa       = (const float*)d_in[10];
  const float* ba       = (const float*)d_in[11];
  const float* Wl       = (const float*)d_in[12];
  const float* bl       = (const float*)d_in[13];
  const int*   edges    = (const int*)d_in[14];

  float* Hout = (float*)d_out;                       // (N, 64)
  float* yhat = Hout + (size_t)N_NODES * OUT_F;      // (N, 2)

  // Workspace layout (floats), ~44 MB total. All offsets 16B-aligned.
  float* P     = (float*)d_ws;                       // N x DTOT
  float* psrc  = P + (size_t)N_NODES * DTOT;         // N
  float* pdst  = psrc + N_NODES;                     // N
  float* qsrc  = pdst + N_NODES;                     // R*N*HEADS
  float* qdst  = qsrc + (size_t)R_REL * N_NODES * HEADS;
  float* hcat  = qdst + (size_t)R_REL * N_NODES * HEADS;  // N x 192
  float* mbuf  = hcat + (size_t)N_NODES * HCOLS;     // N*HEADS (reused per r)
  float* denom = mbuf + N_NODES * HEADS;             // N*HEADS (reused per r)
  float* signb = denom + N_NODES * HEADS;            // E (reused per r)
  float* ewb   = signb + E_EDGES;                    // E*HEADS (reused per r)

  // 1) fused dense projection GEMM (WMMA f32)
  proj_gemm_kernel<<<dim3(DTOT / 64, N_NODES / 16), 128, 0, stream>>>(
      features, Wd, bd, Ww, bw, P);

  // 2) per-node scalars
  pq_kernel<<<N_NODES / 8, 256, 0, stream>>>(P, Wf, Wa, psrc, pdst, qsrc, qdst);

  // 3) zero hcat accumulator
  zero_kernel<<<(N_NODES * HCOLS + 255) / 256, 256, 0, stream>>>(
      hcat, N_NODES * HCOLS);

  // 4) per-relation attention (scratch reused; stream order serializes)
  for (int r = 0; r < R_REL; ++r) {
    const int* srcp = edges + (size_t)r * 2 * E_EDGES;
    const int* dstp = srcp + E_EDGES;
    const float* qs = qsrc + (size_t)r * N_NODES * HEADS;
    const float* qd = qdst + (size_t)r * N_NODES * HEADS;

    init_md_kernel<<<N_NODES * HEADS / 256, 256, 0, stream>>>(mbuf, denom);
    edge_alpha_kernel<<<E_EDGES / 256, 256, 0, stream>>>(
        srcp, dstp, psrc, pdst, qs, qd, bf, ba + r, signb, ewb, mbuf);
    m_fix_kernel<<<N_NODES * HEADS / 256, 256, 0, stream>>>(mbuf);
    edge_exp_kernel<<<E_EDGES / 256, 256, 0, stream>>>(dstp, ewb, mbuf, denom);
    edge_scatter_kernel<<<E_EDGES / 8, 256, 0, stream>>>(
        srcp, dstp, signb, ewb, denom, P, D_F * (r + 1), hcat, r * HD);
  }

  // 5) HBM-dominant fused T normalize + T@labels (64KB dynamic LDS for labels)
  y_hat_kernel<<<N_NODES, 256, 2 * N_NODES * sizeof(float), stream>>>(
      bi_adj, adj_mask, labels, yhat);

  // 6) final GEMM (WMMA f32)
  out_gemm_kernel<<<N_NODES / 16, 128, 0, stream>>>(hcat, Wl, bl, Hout);
}